// P2G_29798483099832
// MI455X (gfx1250) — compile-verified
//
#include <hip/hip_runtime.h>

// P2G quadratic-B-spline scatter for MI455X (gfx1250, wave32).
// Bottleneck = L2 atomic-add throughput (54M f32 atomics into 16MB of grids,
// fully L2-resident on the 192MB L2). No matrix contraction -> no WMMA path;
// the CDNA5-correct instructions are global_load_b96 (particle pos) and
// no-return global_atomic_add_f32 (STOREcnt-tracked scatter).

#define GRID_N   64
#define G3       (64 * 64 * 64)
#define INV_DX   64.0f
#define EPSCLIP  1e-5f
#define EPSDIV   1e-7f

// ---- zero both accumulators (graph-capture safe, no memset) ---------------
__global__ void p2g_zero(float4* __restrict__ a, float4* __restrict__ b, int n4) {
    int i = blockIdx.x * blockDim.x + threadIdx.x;
    if (i < n4) {
        float4 z; z.x = 0.f; z.y = 0.f; z.z = 0.f; z.w = 0.f;
        a[i] = z;
        b[i] = z;
    }
}

// no-return f32 atomic add, device scope, relaxed: should select the native
// global_atomic_add_f32 (no RTN) on gfx1250.
__device__ __forceinline__ void gatomic_add(float* p, float v) {
    __hip_atomic_fetch_add(p, v, __ATOMIC_RELAXED, __HIP_MEMORY_SCOPE_AGENT);
}

struct __align__(4) f3 { float x, y, z; };

// ---- scatter --------------------------------------------------------------
__global__ void __launch_bounds__(256)
p2g_scatter(const float* __restrict__ pos,
            const float* __restrict__ prob,
            const int*   __restrict__ bidx,
            float*       __restrict__ wprob,   // d_out accumulator
            float*       __restrict__ wgt,     // d_ws  accumulator
            int L) {
    int i = blockIdx.x * blockDim.x + threadIdx.x;
    if (i >= L) return;

    // 12-byte contiguous position load -> global_load_b96
    f3 p = ((const f3*)pos)[i];

    float px = fminf(fmaxf(p.x, EPSCLIP), 1.0f - EPSCLIP);
    float py = fminf(fmaxf(p.y, EPSCLIP), 1.0f - EPSCLIP);
    float pz = fminf(fmaxf(p.z, EPSCLIP), 1.0f - EPSCLIP);

    float xp = px * INV_DX, yp = py * INV_DX, zp = pz * INV_DX;
    int bx = (int)xp, by = (int)yp, bz = (int)zp;   // floor (positive)
    float fx = xp - (float)bx, fy = yp - (float)by, fz = zp - (float)bz;

    float wx[3], wy[3], wz[3];
    wx[0] = 0.5f * (1.0f - fx) * (1.0f - fx);
    wx[1] = 0.75f - (fx - 0.5f) * (fx - 0.5f);
    wx[2] = 0.5f * fx * fx;
    wy[0] = 0.5f * (1.0f - fy) * (1.0f - fy);
    wy[1] = 0.75f - (fy - 0.5f) * (fy - 0.5f);
    wy[2] = 0.5f * fy * fy;
    wz[0] = 0.5f * (1.0f - fz) * (1.0f - fz);
    wz[1] = 0.75f - (fz - 0.5f) * (fz - 0.5f);
    wz[2] = 0.5f * fz * fz;

    float pr   = prob[i];
    int   bofs = bidx[i] * G3;

    // clip guarantees bx,by,bz in [0,63] -> taps in [-1,64]; one unsigned
    // compare per axis is the full bounds check.
#pragma unroll
    for (int dx = 0; dx < 3; ++dx) {
        int tx = bx + dx - 1;
        if ((unsigned)tx >= (unsigned)GRID_N) continue;
#pragma unroll
        for (int dy = 0; dy < 3; ++dy) {
            int ty = by + dy - 1;
            if ((unsigned)ty >= (unsigned)GRID_N) continue;
            float wxy = wx[dx] * wy[dy];
            int rowbase = bofs + (((tx << 6) + ty) << 6) + bz;  // z = bz + (dz-1)
#pragma unroll
            for (int dz = 0; dz < 3; ++dz) {
                int tz = bz + dz - 1;
                if ((unsigned)tz >= (unsigned)GRID_N) continue;
                float w   = wxy * wz[dz];
                int   idx = rowbase + dz - 1;
                gatomic_add(&wgt[idx],   w);        // same L2 sector pair
                gatomic_add(&wprob[idx], w * pr);
            }
        }
    }
}

// ---- finalize: out = wprob / (weight + eps), in place, vectorized ---------
__global__ void p2g_finalize(float4* __restrict__ out,
                             const float4* __restrict__ wgt, int n4) {
    int i = blockIdx.x * blockDim.x + threadIdx.x;
    if (i < n4) {
        float4 o = out[i];
        float4 w = wgt[i];
        o.x = o.x / (w.x + EPSDIV);
        o.y = o.y / (w.y + EPSDIV);
        o.z = o.z / (w.z + EPSDIV);
        o.w = o.w / (w.w + EPSDIV);
        out[i] = o;
    }
}

extern "C" void kernel_launch(void* const* d_in, const int* in_sizes, int n_in,
                              void* d_out, int out_size, void* d_ws, size_t ws_size,
                              hipStream_t stream) {
    const float* pos  = (const float*)d_in[0];   // (L,3) f32
    const float* prob = (const float*)d_in[1];   // (L,)  f32
    const int*   bidx = (const int*)  d_in[2];   // (L,)  i32
    float* out = (float*)d_out;                  // B*G3 f32 -> wprob accumulator, then result
    float* wgt = (float*)d_ws;                   // B*G3 f32 weight accumulator (8 MB scratch)

    int L  = in_sizes[1];        // particle count
    int n  = out_size;           // B * 64^3
    int n4 = n >> 2;             // divisible by 4

    const int BLK = 256;         // 8 wave32 per block
    p2g_zero<<<(n4 + BLK - 1) / BLK, BLK, 0, stream>>>((float4*)out, (float4*)wgt, n4);
    p2g_scatter<<<(L + BLK - 1) / BLK, BLK, 0, stream>>>(pos, prob, bidx, out, wgt, L);
    p2g_finalize<<<(n4 + BLK - 1) / BLK, BLK, 0, stream>>>((float4*)out, (const float4*)wgt, n4);
}